// BivariateGaussianKernel_21131239096559
// MI455X (gfx1250) — compile-verified
//
#include <hip/hip_runtime.h>
#include <hip/hip_bf16.h>

typedef __attribute__((ext_vector_type(2))) float v2f;
typedef __attribute__((ext_vector_type(8))) float v8f;

#define EPS_F   1e-07f
#define LOG2E_F 1.4426950408889634f

// ---------------------------------------------------------------------------
// Stage 1: each wave owns a 16-column tile of the (N x M) weight matrix and a
// strided subset of 16-row tiles. The K=2 cross-term inputs@x^T is computed
// with V_WMMA_F32_16X16X4_F32 (K padded to 4 with zeros); the Gaussian weight
// and the two reductions (numerator / denominator) are VALU + v_exp_f32.
// Partial sums are written deterministically to workspace.
// ---------------------------------------------------------------------------
__global__ __launch_bounds__(128) void nw_gauss_stage1(
    const float* __restrict__ xq,    // M x 2 query points
    const float* __restrict__ inp,   // N x 2 data points
    const float* __restrict__ outv,  // N    data values
    const float* __restrict__ bwv,   // M    bandwidths
    float* __restrict__ numw,        // RS x M partial numerators
    float* __restrict__ denw,        // RS x M partial denominators
    int N, int M, int RS)
{
    const int wave = threadIdx.x >> 5;
    const int lane = threadIdx.x & 31;
    const int half = lane >> 4;   // 0: rows r..r+7 & K=0..1 ; 1: rows +8 & K=2..3 (pad)
    const int n    = lane & 15;   // column within the 16-wide tile

    const int numJT = M >> 4;
    const int jt = blockIdx.x * 4 + wave;   // column-tile index (wave-uniform)
    if (jt >= numJT) return;                // wave-uniform exit: EXEC stays all-ones
    const int rs = blockIdx.y;              // row chunk
    const int jbase = jt << 4;
    const int j = jbase + n;

    // Per-lane column constants (loop invariant).
    const float xjx  = xq[2 * j + 0];
    const float xjy  = xq[2 * j + 1];
    const float b2   = xjx * xjx + xjy * xjy;
    const float bw   = bwv[j];
    const float cfac = LOG2E_F / (2.0f * bw * bw);   // exp(-d2/(2bw^2)) = exp2(-d2*cfac)

    // B matrix (K x N = 4 x 16, f32 layout): VGPR0 = K0 (lanes 0-15) / K2 (lanes 16-31),
    // VGPR1 = K1 / K3. K=2,3 are zero padding.
    v2f bmat;
    bmat.x = half ? 0.0f : xjx;
    bmat.y = half ? 0.0f : xjy;

    float numsum = 0.0f;
    float densum = 0.0f;

    const int nTiles = N >> 4;
    const float4* in4  = (const float4*)inp;
    const float4* out4 = (const float4*)outv;

    for (int it = rs; it < nTiles; it += RS) {
        const int ibase = it << 4;

        // A matrix (M x K = 16 x 4): lanes 0-15 hold row ibase+lane (K=0,1);
        // lanes 16-31 hold the K=2,3 zero padding.
        v2f amat;
        if (half == 0) {
            amat.x = inp[2 * (ibase + lane) + 0];
            amat.y = inp[2 * (ibase + lane) + 1];
        } else {
            amat.x = 0.0f;
            amat.y = 0.0f;
        }

        v8f c = {};
        // D[r][n] = dot(inputs[ibase + half*8 + r], x[jbase + n])
        v8f d = __builtin_amdgcn_wmma_f32_16x16x4_f32(
            /*neg_a=*/false, amat, /*neg_b=*/false, bmat,
            /*c_mod=*/(short)0, c, /*reuse_a=*/false, /*reuse_b=*/false);

        // Row-local data for this lane's 8 rows: R .. R+7 (16B-aligned vector loads).
        const int R = ibase + half * 8;
        const float4 p0 = in4[(R >> 1) + 0];
        const float4 p1 = in4[(R >> 1) + 1];
        const float4 p2 = in4[(R >> 1) + 2];
        const float4 p3 = in4[(R >> 1) + 3];
        const float4 o0 = out4[(R >> 2) + 0];
        const float4 o1 = out4[(R >> 2) + 1];

        const float a2v[8] = {
            p0.x * p0.x + p0.y * p0.y, p0.z * p0.z + p0.w * p0.w,
            p1.x * p1.x + p1.y * p1.y, p1.z * p1.z + p1.w * p1.w,
            p2.x * p2.x + p2.y * p2.y, p2.z * p2.z + p2.w * p2.w,
            p3.x * p3.x + p3.y * p3.y, p3.z * p3.z + p3.w * p3.w };
        const float ov[8] = { o0.x, o0.y, o0.z, o0.w, o1.x, o1.y, o1.z, o1.w };

        #pragma unroll
        for (int r = 0; r < 8; ++r) {
            float d2 = fmaf(-2.0f, d[r], a2v[r] + b2);  // a2 + b2 - 2*dot
            d2 = fmaxf(d2, 0.0f);
            const float w = __builtin_amdgcn_exp2f(-d2 * cfac);
            numsum = fmaf(w, ov[r], numsum);
            densum += w;
        }
    }

    // Combine the two row-halves of this column (lane n <-> lane n+16).
    numsum += __shfl_xor(numsum, 16, 32);
    densum += __shfl_xor(densum, 16, 32);

    if (half == 0) {
        numw[(size_t)rs * M + j] = numsum;
        denw[(size_t)rs * M + j] = densum;
    }
}

// ---------------------------------------------------------------------------
// Stage 2: deterministic 8-way partial reduction + normalization.
// ---------------------------------------------------------------------------
__global__ __launch_bounds__(256) void nw_gauss_stage2(
    const float* __restrict__ numw,
    const float* __restrict__ denw,
    float* __restrict__ out,
    int M, int RS)
{
    const int j = blockIdx.x * 256 + threadIdx.x;
    if (j >= M) return;
    float num = 0.0f, den = 0.0f;
    for (int rs = 0; rs < RS; ++rs) {
        num += numw[(size_t)rs * M + j];
        den += denw[(size_t)rs * M + j];
    }
    out[j] = num / (den + EPS_F);
}

extern "C" void kernel_launch(void* const* d_in, const int* in_sizes, int n_in,
                              void* d_out, int out_size, void* d_ws, size_t ws_size,
                              hipStream_t stream) {
    // setup_inputs() order: x (M x 2), inputs (N x 2), outputs (N), bandwidth (M)
    const float* xq   = (const float*)d_in[0];
    const float* inp  = (const float*)d_in[1];
    const float* outv = (const float*)d_in[2];
    const float* bwv  = (const float*)d_in[3];
    float* out = (float*)d_out;

    const int M = in_sizes[0] / 2;
    const int N = in_sizes[1] / 2;

    // Row-split factor: 8 gives 4096 waves; clamp if workspace is small.
    int RS = 8;
    while (RS > 1 && (size_t)2 * RS * (size_t)M * sizeof(float) > ws_size) RS >>= 1;

    float* numw = (float*)d_ws;
    float* denw = numw + (size_t)RS * M;

    const int numJT = M / 16;                 // 16-column tiles
    dim3 grid1((numJT + 3) / 4, RS);          // 4 waves (column tiles) per 128-thread block
    nw_gauss_stage1<<<grid1, 128, 0, stream>>>(xq, inp, outv, bwv, numw, denw, N, M, RS);

    nw_gauss_stage2<<<(M + 255) / 256, 256, 0, stream>>>(numw, denw, out, M, RS);
}